// KMaxPool_28870770163873
// MI455X (gfx1250) — compile-verified
//
#include <hip/hip_runtime.h>
#include <cmath>
#include <climits>

namespace {
constexpr int kS = 4096;           // reduction axis length
constexpr int kK = 8;              // top-k
constexpr int kLanes = 32;         // wave32 on gfx1250
constexpr int kWavesPerBlock = 8;  // 256-thread blocks
constexpr int kBlockDim = kLanes * kWavesPerBlock;
constexpr int kMaxCand = 32;       // candidate buffer cap (>= k; ties headroom)
}

typedef float v4f __attribute__((ext_vector_type(4)));  // native clang vector for NT loads

// Branchless insert of v into a descending sorted 8-list of VALUES:
// t0 = max(t0,v); tj = med3(t[j-1], t[j], v).  8 VALU ops, no indices.
__device__ __forceinline__ void med3_insert8(float (&t)[kK], float v) {
#pragma unroll
  for (int j = kK - 1; j >= 1; --j)
    t[j] = __builtin_amdgcn_fmed3f(t[j - 1], t[j], v);
  t[0] = fmaxf(t[0], v);
}

__device__ __forceinline__ void cmp_exchange(float& a, float& b) {
  float hi = fmaxf(a, b);
  float lo = fminf(a, b);
  a = hi; b = lo;
}

__global__ __launch_bounds__(kBlockDim)
void kmax_pool_kernel(const float* __restrict__ x, float* __restrict__ out, int rows) {
  __shared__ float s_cval[kWavesPerBlock * kMaxCand];
  __shared__ int   s_cidx[kWavesPerBlock * kMaxCand];

  const int wave = threadIdx.x >> 5;
  const int lane = threadIdx.x & (kLanes - 1);
  const int row  = blockIdx.x * kWavesPerBlock + wave;
  if (row >= rows) return;

  const v4f* rowp = reinterpret_cast<const v4f*>(x) + (size_t)row * (kS / 4);

  // ---------------- Pass 1: values-only top-8 per lane (HBM stream) -------
  float t[kK];
#pragma unroll
  for (int j = 0; j < kK; ++j) t[j] = -INFINITY;

#pragma unroll 4
  for (int q = lane; q < kS / 4; q += kLanes) {
    __builtin_prefetch(rowp + q + 8 * kLanes, 0, 1);  // global_prefetch_b8, ~4KiB ahead
    v4f v = rowp[q];
    med3_insert8(t, v[0]);
    med3_insert8(t, v[1]);
    med3_insert8(t, v[2]);
    med3_insert8(t, v[3]);
  }

  // ------------- Wave32 bitonic top-8 merge (values), xor butterfly -------
#pragma unroll
  for (int off = kLanes / 2; off >= 1; off >>= 1) {
    float b[kK];
#pragma unroll
    for (int j = 0; j < kK; ++j)                  // partner's list, reversed
      b[j] = __shfl_xor(t[kK - 1 - j], off, kLanes);
#pragma unroll
    for (int j = 0; j < kK; ++j)                  // bitonic half-clean: top-8 multiset
      t[j] = fmaxf(t[j], b[j]);
    // bitonic cleanup (distance 4, 2, 1) -> sorted descending
    cmp_exchange(t[0], t[4]); cmp_exchange(t[1], t[5]);
    cmp_exchange(t[2], t[6]); cmp_exchange(t[3], t[7]);
    cmp_exchange(t[0], t[2]); cmp_exchange(t[1], t[3]);
    cmp_exchange(t[4], t[6]); cmp_exchange(t[5], t[7]);
    cmp_exchange(t[0], t[1]); cmp_exchange(t[2], t[3]);
    cmp_exchange(t[4], t[5]); cmp_exchange(t[6], t[7]);
  }
  const float t8 = t[kK - 1];   // exact 8th-largest value of the row (all lanes)

  // ---------------- Pass 2: gather indices of v >= t8 (L2-resident) -------
  const unsigned lanemask_lt = (1u << lane) - 1u;
  unsigned cnt = 0;  // wave-uniform candidate count
  float* cval = &s_cval[wave * kMaxCand];
  int*   cidx = &s_cidx[wave * kMaxCand];

  for (int q = lane; q < kS / 4; q += kLanes) {
    v4f v = __builtin_nontemporal_load(&rowp[q]);   // b128 load, NT hint (last use)
    float m4 = fmaxf(fmaxf(v[0], v[1]), fmaxf(v[2], v[3]));
    unsigned any = (unsigned)__ballot(m4 >= t8);
    if (any) {                                  // rare, wave-uniform branch
      int base = q * 4;
#pragma unroll
      for (int c = 0; c < 4; ++c) {
        float vc = v[c];
        bool p = vc >= t8;
        unsigned m = (unsigned)__ballot(p);
        if (m) {
          unsigned pos = cnt + (unsigned)__popc(m & lanemask_lt);
          if (p && pos < (unsigned)kMaxCand) {
            cval[pos] = vc;
            cidx[pos] = base + c;
          }
          cnt += (unsigned)__popc(m);
        }
      }
    }
  }
  if (cnt > (unsigned)kMaxCand) cnt = kMaxCand;

  // -------- Exact top-8 among candidates: (value desc, index asc) ---------
  // >= 8 candidates exist by definition of t8; ties handled by index order.
  float cv; int ci0;
  if (lane < (int)cnt) { cv = cval[lane]; ci0 = cidx[lane]; }
  else                 { cv = -INFINITY;  ci0 = INT_MAX;    }

  float outv[kK]; int outi[kK];
#pragma unroll
  for (int r = 0; r < kK; ++r) {
    float c = cv; int ci = ci0;
#pragma unroll
    for (int off = kLanes / 2; off >= 1; off >>= 1) {
      float ov = __shfl_xor(c,  off, kLanes);
      int   oi = __shfl_xor(ci, off, kLanes);
      if (ov > c || (ov == c && oi < ci)) { c = ov; ci = oi; }
    }
    if (ci0 == ci) { cv = -INFINITY; ci0 = INT_MAX; }  // winner retires
    outv[r] = c; outi[r] = ci;
  }

  // --------- Restore original relative order: sort 8 by index asc ---------
#pragma unroll
  for (int a = 0; a < kK - 1; ++a) {
#pragma unroll
    for (int b = 0; b < kK - 1 - a; ++b) {
      if (outi[b] > outi[b + 1]) {
        int   ii = outi[b]; outi[b] = outi[b + 1]; outi[b + 1] = ii;
        float fv = outv[b]; outv[b] = outv[b + 1]; outv[b + 1] = fv;
      }
    }
  }

  if (lane == 0) {
    float4* op = reinterpret_cast<float4*>(out + (size_t)row * kK);
    op[0] = make_float4(outv[0], outv[1], outv[2], outv[3]);
    op[1] = make_float4(outv[4], outv[5], outv[6], outv[7]);
  }
}

extern "C" void kernel_launch(void* const* d_in, const int* in_sizes, int n_in,
                              void* d_out, int out_size, void* d_ws, size_t ws_size,
                              hipStream_t stream) {
  (void)n_in; (void)d_ws; (void)ws_size; (void)out_size;
  const float* x = (const float*)d_in[0];
  float* out = (float*)d_out;
  const int rows = in_sizes[0] / kS;  // 16 * 1024 = 16384
  const int blocks = (rows + kWavesPerBlock - 1) / kWavesPerBlock;
  kmax_pool_kernel<<<blocks, kBlockDim, 0, stream>>>(x, out, rows);
}